// IndustryEncoder_32787780337875
// MI455X (gfx1250) — compile-verified
//
#include <hip/hip_runtime.h>

typedef __attribute__((ext_vector_type(2))) float v2f;
typedef __attribute__((ext_vector_type(4))) float v4f;
typedef __attribute__((ext_vector_type(8))) float v8f;

#define NUM_IND 1000
#define TILES ((NUM_IND + 15) / 16)   // 63 tiles of 16 industries

// ---------------------------------------------------------------------------
// Kernel 1: per-industry table precompute via V_WMMA_F32_16X16X4_F32.
//   table[i][:] = relu(vars[i] @ W1 + b1) @ W2 + b2 + 0.1*emb[i]
// One wave32 per 16-row tile; EXEC all-ones as WMMA requires.
// ---------------------------------------------------------------------------
__global__ __launch_bounds__(32) void industry_table_wmma(
    const float* __restrict__ vars,   // [1000, 8]
    const float* __restrict__ W1,     // [8, 16]
    const float* __restrict__ b1,     // [16]
    const float* __restrict__ W2,     // [16, 32]
    const float* __restrict__ b2,     // [32]
    const float* __restrict__ emb,    // [1000, 32]
    float* __restrict__ table)        // [1000, 32]
{
    __shared__ float H[16][17];       // padded stride: conflict-free transpose reads

    const int lane = threadIdx.x;     // 0..31
    const int half = lane >> 4;       // 0: K={0,1}-side, 1: K={2,3}-side (A/B layout)
    const int lid  = lane & 15;       // M (for A) / N (for B) index
    const int row0 = blockIdx.x * 16;

    // ---- Layer 1: X[16x8] @ W1[8x16]  (two K=4 WMMA steps) ----
    int xrow = row0 + lid;
    if (xrow > NUM_IND - 1) xrow = NUM_IND - 1;   // clamp (stores guarded later)

    v2f a0, a1, w1b0, w1b1;
    // A-layout 16x4 f32: lanes 0-15 -> K=0,1 ; lanes 16-31 -> K=2,3 ; M=lid
    a0.x = vars[xrow * 8 + (2 * half + 0)];
    a0.y = vars[xrow * 8 + (2 * half + 1)];
    a1.x = vars[xrow * 8 + (4 + 2 * half + 0)];
    a1.y = vars[xrow * 8 + (4 + 2 * half + 1)];
    // B-layout 4x16 f32: same K split across lane halves; N=lid
    w1b0.x = W1[(2 * half + 0) * 16 + lid];
    w1b0.y = W1[(2 * half + 1) * 16 + lid];
    w1b1.x = W1[(4 + 2 * half + 0) * 16 + lid];
    w1b1.y = W1[(4 + 2 * half + 1) * 16 + lid];

    v8f acc = {};
    acc = __builtin_amdgcn_wmma_f32_16x16x4_f32(false, a0, false, w1b0,
                                                (short)0, acc, false, false);
    acc = __builtin_amdgcn_wmma_f32_16x16x4_f32(false, a1, false, w1b1,
                                                (short)0, acc, false, false);

    // bias + ReLU, stage H into LDS row-major.
    // D-layout: VGPR r holds row (r + 8*half), column lid.
    const float bias1 = b1[lid];
#pragma unroll
    for (int r = 0; r < 8; ++r) {
        float h = acc[r] + bias1;
        H[r + 8 * half][lid] = h > 0.0f ? h : 0.0f;
    }
    __syncthreads();

    // ---- Layer 2: H[16x16] @ W2[16x32]  (2 N-tiles x 4 K-chunks) ----
#pragma unroll
    for (int t = 0; t < 2; ++t) {
        v8f acc2 = {};
#pragma unroll
        for (int kc = 0; kc < 4; ++kc) {
            const int k0 = 4 * kc + 2 * half;
            v2f ah, w2b;
            ah.x  = H[lid][k0 + 0];                    // A: M=lid, K per lane-half
            ah.y  = H[lid][k0 + 1];
            w2b.x = W2[(k0 + 0) * 32 + 16 * t + lid];  // B: N = 16*t + lid
            w2b.y = W2[(k0 + 1) * 32 + 16 * t + lid];
            acc2 = __builtin_amdgcn_wmma_f32_16x16x4_f32(false, ah, false, w2b,
                                                         (short)0, acc2, false, false);
        }
        const int n = 16 * t + lid;
        const float bias2 = b2[n];
#pragma unroll
        for (int r = 0; r < 8; ++r) {
            const int row = row0 + r + 8 * half;
            if (row < NUM_IND) {
                table[row * 32 + n] = acc2[r] + bias2 + 0.1f * emb[row * 32 + n];
            }
        }
    }
}

// ---------------------------------------------------------------------------
// Kernel 2: bandwidth-floor gather. 8 lanes per example -> each wave32 store
// is 512B contiguous (4 full 128B lines). Output is streamed with NT stores
// (256MB output > 192MB L2); table reads stay cached (128KB, L2-resident).
// ---------------------------------------------------------------------------
__global__ __launch_bounds__(256) void industry_gather(
    const int* __restrict__ idx,
    const v4f* __restrict__ table4,   // [1000 * 8]
    v4f* __restrict__ out4,           // [BATCH * 8]
    int batch)
{
    const long long t = (long long)blockIdx.x * blockDim.x + threadIdx.x;
    const int b = (int)(t >> 3);
    const int j = (int)(t & 7);
    if (b < batch) {
        const int ind = idx[b];                 // L0/L2 hit (8 lanes share it)
        const v4f v = table4[ind * 8 + j];      // L2-resident random read
        __builtin_nontemporal_store(v, out4 + (long long)b * 8 + j);
    }
}

// ---------------------------------------------------------------------------
// Fallback (only if workspace can't hold the 128KB table): fused per-example.
// ---------------------------------------------------------------------------
__global__ __launch_bounds__(256) void industry_fused_fallback(
    const int* __restrict__ idx, const float* __restrict__ vars,
    const float* __restrict__ W1, const float* __restrict__ b1,
    const float* __restrict__ W2, const float* __restrict__ b2,
    const float* __restrict__ emb, float* __restrict__ out, int batch)
{
    const int b = blockIdx.x * blockDim.x + threadIdx.x;
    if (b >= batch) return;
    const int ind = idx[b];
    float x[8], h[16];
#pragma unroll
    for (int i = 0; i < 8; ++i) x[i] = vars[ind * 8 + i];
#pragma unroll
    for (int n = 0; n < 16; ++n) {
        float s = b1[n];
#pragma unroll
        for (int k = 0; k < 8; ++k) s += x[k] * W1[k * 16 + n];
        h[n] = s > 0.0f ? s : 0.0f;
    }
#pragma unroll
    for (int n = 0; n < 32; ++n) {
        float s = b2[n];
#pragma unroll
        for (int k = 0; k < 16; ++k) s += h[k] * W2[k * 32 + n];
        out[(size_t)b * 32 + n] = s + 0.1f * emb[ind * 32 + n];
    }
}

extern "C" void kernel_launch(void* const* d_in, const int* in_sizes, int n_in,
                              void* d_out, int out_size, void* d_ws, size_t ws_size,
                              hipStream_t stream) {
    const int*   idx  = (const int*)d_in[0];
    const float* vars = (const float*)d_in[1];
    const float* W1   = (const float*)d_in[2];
    const float* b1   = (const float*)d_in[3];
    const float* W2   = (const float*)d_in[4];
    const float* b2   = (const float*)d_in[5];
    const float* emb  = (const float*)d_in[6];
    float*       out  = (float*)d_out;
    const int batch   = in_sizes[0];

    const size_t table_bytes = (size_t)NUM_IND * 32 * sizeof(float); // 128 KB
    if (ws_size >= table_bytes) {
        float* table = (float*)d_ws;
        industry_table_wmma<<<TILES, 32, 0, stream>>>(vars, W1, b1, W2, b2, emb, table);
        const long long threads = (long long)batch * 8;
        const int blocks = (int)((threads + 255) / 256);
        industry_gather<<<blocks, 256, 0, stream>>>(idx, (const v4f*)table,
                                                    (v4f*)out, batch);
    } else {
        industry_fused_fallback<<<(batch + 255) / 256, 256, 0, stream>>>(
            idx, vars, W1, b1, W2, b2, emb, out, batch);
    }
}